// ACMGraphConvolution_16604343566778
// MI455X (gfx1250) — compile-verified
//
#include <hip/hip_runtime.h>
#include <math.h>

#define N_NODES 50000
#define F_IN    512
#define F_OUT   128
#define NE      800000

#define KTILES   (F_IN / 8)      // 64 k-pairs (2 wmma steps each)
#define CTILES   (F_OUT / 16)    // 8 column tiles
#define WPACKN   (CTILES * KTILES * 32 * 4)   // 65536 floats per packed W

typedef __attribute__((ext_vector_type(2))) float v2f;
typedef __attribute__((ext_vector_type(8))) float v8f;

#define LDS_STRIDE 516  // 512 + 4-float pad: lane m hits banks {4m+k,4m+k+1}, conflict-free b64 reads

// ---------------------------------------------------------------------------
// One-time weight repack into WMMA B-fragment order.
// packed[w][tile t][kpair p][lane l] = float4{ W[k][c], W[k+1][c], W[k+4][c], W[k+5][c] }
//   where c = t*16 + (l&15), k = 8p + 2*(l>>4)
// So the GEMM's B fragment for steps k0=8p and k0+4 is ONE coalesced b128 load.
// ---------------------------------------------------------------------------
__global__ __launch_bounds__(256) void repack_w(
    const float* __restrict__ wl,
    const float* __restrict__ wh,
    const float* __restrict__ wm,
    float* __restrict__ packed)       // 3 * WPACKN floats
{
    const int tid = blockIdx.x * 256 + threadIdx.x;     // one float4 per thread
    const int w   = tid >> 14;                          // /16384 float4s per matrix
    const int rem = tid & 16383;
    const int t   = rem >> 11;                          // /2048
    const int r2  = rem & 2047;
    const int p   = r2 >> 5;
    const int l   = r2 & 31;

    const float* W = (w == 0) ? wl : (w == 1) ? wh : wm;
    const int c = t * 16 + (l & 15);
    const int k = 8 * p + ((l >> 4) << 1);

    float4 v;
    v.x = W[(k + 0) * F_OUT + c];
    v.y = W[(k + 1) * F_OUT + c];
    v.z = W[(k + 4) * F_OUT + c];
    v.w = W[(k + 5) * F_OUT + c];
    ((float4*)packed)[tid] = v;
}

// ---------------------------------------------------------------------------
// Fused triple GEMM: h_low = x@Wl, h_high = x@Wh, o_mlp = relu(x@Wm)
// One workgroup (8 waves) per 16-row strip; wave w owns columns [16w,16w+16).
// fp32 WMMA (V_WMMA_F32_16X16X4_F32) keeps full precision; GEMM is far from
// the HBM/atomic bottleneck on MI455X, so fp32 is the right precision.
// Per 2 K-steps: 3x global_load_b128 (packed B) + ds_load_2addr_b64 (A) + 6 wmma.
// ---------------------------------------------------------------------------
__global__ __launch_bounds__(256) void gemm3_wmma_f32(
    const float* __restrict__ x,
    const float* __restrict__ packed,   // repacked Wl|Wh|Wm
    float* __restrict__ h_low,
    float* __restrict__ h_high,
    float* __restrict__ o_mlp)
{
    __shared__ float lds[16 * LDS_STRIDE];
    const int t    = threadIdx.x;
    const int row0 = blockIdx.x * 16;

    // Cooperative stage of A tile (16 x 512 f32) into LDS, b128 loads.
#pragma unroll
    for (int i = 0; i < 8; ++i) {
        int idx = t + i * 256;          // float4 index, 0..2047
        int r   = idx >> 7;             // 128 float4 per row
        int c4  = idx & 127;
        const float4 v = *(const float4*)(x + (size_t)(row0 + r) * F_IN + c4 * 4);
        *(float4*)(&lds[r * LDS_STRIDE + c4 * 4]) = v;
    }
    __syncthreads();

    const int lane    = t & 31;
    const int wave    = t >> 5;           // 0..7 -> column tile
    const int m       = lane & 15;
    const int khalf   = (lane >> 4) << 1; // 0 or 2
    const int colbase = wave * 16;

    // per-wave packed-B base (float4 granularity), fully coalesced
    const float4* pL = (const float4*)(packed) + (size_t)wave * KTILES * 32 + lane;
    const float4* pH = pL + (WPACKN / 4);
    const float4* pM = pH + (WPACKN / 4);

    v8f accL = {}; v8f accH = {}; v8f accM = {};

#pragma unroll 4
    for (int p = 0; p < KTILES; ++p) {
        const int k0 = 8 * p;
        // A fragments for steps k0 and k0+4 (merged into ds_load_2addr_b64)
        v2f a0 = *(const v2f*)(&lds[m * LDS_STRIDE + k0 + khalf]);
        v2f a1 = *(const v2f*)(&lds[m * LDS_STRIDE + k0 + 4 + khalf]);

        const float4 BL = pL[(size_t)p * 32];
        const float4 BH = pH[(size_t)p * 32];
        const float4 BM = pM[(size_t)p * 32];

        v2f b;
        b.x = BL.x; b.y = BL.y;
        accL = __builtin_amdgcn_wmma_f32_16x16x4_f32(false, a0, false, b, (short)0, accL, false, false);
        b.x = BH.x; b.y = BH.y;
        accH = __builtin_amdgcn_wmma_f32_16x16x4_f32(false, a0, false, b, (short)0, accH, false, false);
        b.x = BM.x; b.y = BM.y;
        accM = __builtin_amdgcn_wmma_f32_16x16x4_f32(false, a0, false, b, (short)0, accM, false, false);
        b.x = BL.z; b.y = BL.w;
        accL = __builtin_amdgcn_wmma_f32_16x16x4_f32(false, a1, false, b, (short)0, accL, false, false);
        b.x = BH.z; b.y = BH.w;
        accH = __builtin_amdgcn_wmma_f32_16x16x4_f32(false, a1, false, b, (short)0, accH, false, false);
        b.x = BM.z; b.y = BM.w;
        accM = __builtin_amdgcn_wmma_f32_16x16x4_f32(false, a1, false, b, (short)0, accM, false, false);
    }

    // C/D layout: VGPR v -> row (v + 8*(lane>>4)), col (lane&15)
    const int mbase = (lane >> 4) << 3;
#pragma unroll
    for (int v = 0; v < 8; ++v) {
        const size_t off = (size_t)(row0 + mbase + v) * F_OUT + colbase + m;
        h_low[off]  = accL[v];
        h_high[off] = accH[v];
        o_mlp[off]  = fmaxf(accM[v], 0.0f);   // relu fused for mlp branch
    }
}

// ---------------------------------------------------------------------------
// SpMM scatter: one wave per edge; lane l handles cols [4l,4l+4).
// out[row_e] += val_e * h[col_e]   (non-returning global_atomic_add_f32)
// Edge scalars go through readfirstlane -> scalar loads (uniform per wave).
// ---------------------------------------------------------------------------
__global__ __launch_bounds__(256) void spmm_scatter(
    const int*   __restrict__ row,
    const int*   __restrict__ col,
    const float* __restrict__ val,
    const float* __restrict__ h,
    float*       __restrict__ out)
{
    const int lane = threadIdx.x & 31;
    const int e    = __builtin_amdgcn_readfirstlane(blockIdx.x * 8 + (threadIdx.x >> 5));
    const int r    = row[e];
    const int c    = col[e];
    const float v  = val[e];

    const float4 hv = *(const float4*)(h + (size_t)c * F_OUT + lane * 4);
    float* dst = out + (size_t)r * F_OUT + lane * 4;
    atomicAdd(dst + 0, v * hv.x);
    atomicAdd(dst + 1, v * hv.y);
    atomicAdd(dst + 2, v * hv.z);
    atomicAdd(dst + 3, v * hv.w);
}

// ---------------------------------------------------------------------------
// Attention epilogue: one wave per node row.
// ---------------------------------------------------------------------------
__global__ __launch_bounds__(256) void attention_combine(
    const float* __restrict__ s_low,
    const float* __restrict__ s_high,
    const float* __restrict__ o_mlp,
    const float* __restrict__ av_low,
    const float* __restrict__ av_high,
    const float* __restrict__ av_mlp,
    const float* __restrict__ att,     // [3,3] row-major
    float*       __restrict__ out)
{
    const int lane = threadIdx.x & 31;
    const int i    = blockIdx.x * 8 + (threadIdx.x >> 5);
    const size_t base = (size_t)i * F_OUT + lane * 4;

    float4 ol = *(const float4*)(s_low  + base);
    float4 oh = *(const float4*)(s_high + base);
    float4 om = *(const float4*)(o_mlp  + base);
    ol.x = fmaxf(ol.x, 0.f); ol.y = fmaxf(ol.y, 0.f); ol.z = fmaxf(ol.z, 0.f); ol.w = fmaxf(ol.w, 0.f);
    oh.x = fmaxf(oh.x, 0.f); oh.y = fmaxf(oh.y, 0.f); oh.z = fmaxf(oh.z, 0.f); oh.w = fmaxf(oh.w, 0.f);

    const float4 al = *(const float4*)(av_low  + lane * 4);
    const float4 ah = *(const float4*)(av_high + lane * 4);
    const float4 am = *(const float4*)(av_mlp  + lane * 4);

    float dl = ol.x*al.x + ol.y*al.y + ol.z*al.z + ol.w*al.w;
    float dh = oh.x*ah.x + oh.y*ah.y + oh.z*ah.z + oh.w*ah.w;
    float dm = om.x*am.x + om.y*am.y + om.z*am.z + om.w*am.w;

    // wave32 all-reduce
#pragma unroll
    for (int off = 16; off >= 1; off >>= 1) {
        dl += __shfl_xor(dl, off);
        dh += __shfl_xor(dh, off);
        dm += __shfl_xor(dm, off);
    }

    const float sl = 1.0f / (1.0f + expf(-dl));
    const float sh = 1.0f / (1.0f + expf(-dh));
    const float sm = 1.0f / (1.0f + expf(-dm));

    float z0 = (sl * att[0] + sh * att[3] + sm * att[6]) * (1.0f / 3.0f);
    float z1 = (sl * att[1] + sh * att[4] + sm * att[7]) * (1.0f / 3.0f);
    float z2 = (sl * att[2] + sh * att[5] + sm * att[8]) * (1.0f / 3.0f);

    const float zm = fmaxf(z0, fmaxf(z1, z2));
    const float e0 = expf(z0 - zm), e1 = expf(z1 - zm), e2 = expf(z2 - zm);
    const float inv = 1.0f / (e0 + e1 + e2);
    const float a0 = 3.0f * e0 * inv, a1 = 3.0f * e1 * inv, a2 = 3.0f * e2 * inv;

    float4 o;
    o.x = a0 * ol.x + a1 * oh.x + a2 * om.x;
    o.y = a0 * ol.y + a1 * oh.y + a2 * om.y;
    o.z = a0 * ol.z + a1 * oh.z + a2 * om.z;
    o.w = a0 * ol.w + a1 * oh.w + a2 * om.w;
    *(float4*)(out + base) = o;
}

// ---------------------------------------------------------------------------
extern "C" void kernel_launch(void* const* d_in, const int* in_sizes, int n_in,
                              void* d_out, int out_size, void* d_ws, size_t ws_size,
                              hipStream_t stream)
{
    const float* x        = (const float*)d_in[0];
    const int*   row_low  = (const int*)  d_in[1];
    const int*   col_low  = (const int*)  d_in[2];
    const float* val_low  = (const float*)d_in[3];
    const int*   row_high = (const int*)  d_in[4];
    const int*   col_high = (const int*)  d_in[5];
    const float* val_high = (const float*)d_in[6];
    const float* w_low    = (const float*)d_in[7];
    const float* w_high   = (const float*)d_in[8];
    const float* w_mlp    = (const float*)d_in[9];
    const float* av_low   = (const float*)d_in[10];
    const float* av_high  = (const float*)d_in[11];
    const float* av_mlp   = (const float*)d_in[12];
    const float* att_vec  = (const float*)d_in[13];

    const size_t mat = (size_t)N_NODES * F_OUT;   // 6.4M floats
    float* h_low  = (float*)d_ws;                 // x@Wl (pre-relu)
    float* h_high = h_low  + mat;                 // x@Wh (pre-relu)
    float* o_mlp  = h_high + mat;                 // relu(x@Wm)
    float* s_low  = o_mlp  + mat;                 // spmm result (low)
    float* s_high = h_low;                        // alias: h_low dead after spmm_low
    float* packed = s_low  + mat;                 // 3 * 65536 floats (0.79 MB)

    // 0) repack weights into WMMA B-fragment lane order (one b128 per 2 k-steps)
    repack_w<<<(3 * WPACKN / 4) / 256, 256, 0, stream>>>(w_low, w_high, w_mlp, packed);

    // 1) fused triple GEMM (WMMA f32)
    gemm3_wmma_f32<<<N_NODES / 16, 256, 0, stream>>>(x, packed, h_low, h_high, o_mlp);

    // 2) spmm low:  s_low = A_low @ h_low
    hipMemsetAsync(s_low, 0, mat * sizeof(float), stream);
    spmm_scatter<<<NE / 8, 256, 0, stream>>>(row_low, col_low, val_low, h_low, s_low);

    // 3) spmm high: s_high = A_high @ h_high   (reuses h_low's buffer)
    hipMemsetAsync(s_high, 0, mat * sizeof(float), stream);
    spmm_scatter<<<NE / 8, 256, 0, stream>>>(row_high, col_high, val_high, h_high, s_high);

    // 4) attention + blend
    attention_combine<<<N_NODES / 8, 256, 0, stream>>>(s_low, s_high, o_mlp,
                                                       av_low, av_high, av_mlp,
                                                       att_vec, (float*)d_out);
}